// MultiHeadAttention_47330539602717
// MI455X (gfx1250) — compile-verified
//
#include <hip/hip_runtime.h>
#include <hip/hip_bf16.h>
#include <stdint.h>

typedef __attribute__((ext_vector_type(2))) float v2f;
typedef __attribute__((ext_vector_type(8))) float v8f;

// Problem constants (from reference): B=8, Lq=128, S=4096, d_model=1024, h=16, d_k=64
#define NBATCH   8
#define LQ       128
#define SRC      4096
#define DMODEL   1024
#define NHEAD    16
#define DK       64
#define BH       128          // NBATCH*NHEAD
#define SMOOTHF  9.0f

// LDS pitches (64 banks x 4B)
#define IP   68    // i_s   [LQ][DK]  pitch: natural layout, rows 16B-aligned, bank=(4l+d)%64
#define VSP  36    // vstage[32][32]  pitch (16B-aligned rows, conflict-free A reads)
#define WSP  64    // wstage[32][64]  pitch
#define PP   68    // vp_s  [32][64]  pitch (aliases staging)
#define ZP   129   // z_s   [32][128] pitch: bank=(j+l)%64, conflict-free column reads

static __device__ __forceinline__ v8f wmma_f32(v2f a, v2f b, v8f c) {
    // D = A(16x4,f32) * B(4x16,f32) + C(16x16,f32)
    return __builtin_amdgcn_wmma_f32_16x16x4_f32(false, a, false, b, (short)0, c,
                                                 false, false);
}

// gfx1250 async global->LDS copy (ASYNCcnt-tracked, no VGPR round-trip)
static __device__ __forceinline__ void async_g2l_b128(void* lds_ptr, const void* gptr) {
    const unsigned lds_off = (unsigned)(uintptr_t)lds_ptr;  // LDS aperture: offset in addr[31:0]
    asm volatile("global_load_async_to_lds_b128 %0, %1, off"
                 :: "v"(lds_off), "v"(gptr)
                 : "memory");
}
static __device__ __forceinline__ void wait_async0() {
    asm volatile("s_wait_asynccnt 0x0" ::: "memory");
}

// One workgroup (256 threads = 8 waves) per batch-head.
// Fuses: vp-slice = v[batch] @ Wv[:,head]  ->  logits = vp @ i^T  ->  softmax(q)
//        -> exp(9p) streaming softmax over S with mod-128 bucketing -> y[bh][0:64]
__global__ __launch_bounds__(256) void attn_fused_kernel(
    const float* __restrict__ v,   // (8,4096,1024)
    const float* __restrict__ ii,  // (128,128,64)
    const float* __restrict__ Wv,  // (1024,1024)
    const float* __restrict__ bv,  // (1024,)
    float* __restrict__ y)         // (128,64)  == x_sum viewed as (8,1024)
{
    __shared__ float i_s[LQ * IP];                 // i[l][d], natural layout
    __shared__ float stage[32 * VSP + 32 * WSP];   // staging; later aliased as vp_s
    __shared__ float z_s[32 * ZP];                 // logits / E2 block [s_local][l]
    __shared__ float Z2[LQ];                       // softmax2 denominators per query l
    __shared__ float w_s[LQ];                      // final weights per query bucket

    float* vstage = stage;                 // [32][VSP]
    float* wstage = stage + 32 * VSP;      // [32][WSP]
    float* vp_s   = stage;                 // [32][PP]  (reused after staging done)

    const int t     = threadIdx.x;
    const int bh    = blockIdx.x;          // 0..127
    const int batch = bh >> 4;
    const int head  = bh & 15;
    const int wave  = t >> 5;
    const int lane  = t & 31;
    const int lo    = lane & 15;
    const int hi    = lane >> 4;

    // ---- async-load i[bh] into LDS, natural [l][d] layout (8 x b128 per thread)
#pragma unroll
    for (int it = 0; it < 8; ++it) {
        const int idx = t + it * 256;            // float4 index, 0..2047
        const int l = idx >> 4, d4 = (idx & 15) * 4;
        async_g2l_b128(&i_s[l * IP + d4], &ii[(size_t)bh * (LQ * DK) + idx * 4]);
    }
    if (t < LQ) { Z2[t] = 0.0f; w_s[t] = 0.0f; }

    // per-thread bucket accumulators: thread owns (l_own = t&127, q in [qhalf*64, +64))
    float nb[64];
#pragma unroll
    for (int k = 0; k < 64; ++k) nb[k] = 0.0f;
    const int l_own = t & 127;
    const int qhalf = t >> 7;

    // GEMM1 wave tiling: 32x64 output -> 2 M-tiles x 4 N-tiles, one per wave
    const int mtg = wave & 1;
    const int ntg = wave >> 1;
    const float bvv = bv[head * DK + ntg * 16 + lo];  // bias for this wave's N col

    wait_async0();
    __syncthreads();

    for (int blk = 0; blk < 32; ++blk) {
        const int s0 = blk * 32;

        // ================= GEMM1 slice: vp_blk(32x64) = v_blk(32x1024) @ Wv_slice
        v8f cg = {0.f, 0.f, 0.f, 0.f, 0.f, 0.f, 0.f, 0.f};
        for (int kc = 0; kc < DMODEL; kc += 32) {
            __syncthreads();  // previous readers of staging done
            {   // stage v block: 32 rows x 32 cols, one async b128/thread
                const int r = t >> 3, c4 = (t & 7) * 4;
                const float* gp = &v[((size_t)(batch * SRC + s0 + r)) * DMODEL + kc + c4];
                async_g2l_b128(&vstage[r * VSP + c4], gp);
                if (kc + 32 < DMODEL)  // gfx1250 global_prefetch_b8
                    __builtin_prefetch(gp + 32, 0, 1);
            }
            {   // stage Wv chunk: 32 rows x 64 cols, two async b128/thread
                const int r = t >> 4, c4 = (t & 15) * 4;
                async_g2l_b128(&wstage[r * WSP + c4],
                               &Wv[(size_t)(kc + r) * DMODEL + head * DK + c4]);
                async_g2l_b128(&wstage[(r + 16) * WSP + c4],
                               &Wv[(size_t)(kc + r + 16) * DMODEL + head * DK + c4]);
            }
            wait_async0();
            __syncthreads();
#pragma unroll
            for (int kk = 0; kk < 8; ++kk) {
                v2f a, b;
                a.x = vstage[(mtg * 16 + lo) * VSP + kk * 4 + 2 * hi + 0];
                a.y = vstage[(mtg * 16 + lo) * VSP + kk * 4 + 2 * hi + 1];
                b.x = wstage[(kk * 4 + 2 * hi + 0) * WSP + ntg * 16 + lo];
                b.y = wstage[(kk * 4 + 2 * hi + 1) * WSP + ntg * 16 + lo];
                cg = wmma_f32(a, b, cg);
            }
        }
        __syncthreads();
        // write vp block (+bv) into LDS (aliases staging)
#pragma unroll
        for (int vv = 0; vv < 8; ++vv)
            vp_s[(mtg * 16 + vv + 8 * hi) * PP + ntg * 16 + lo] = cg[vv] + bvv;
        __syncthreads();

        // ================= logits: z(32x128) = vp_blk(32x64) @ i^T(64x128)
        // B[k=d][n=l] = i[l][d]: lane's two K-values are consecutive d -> ds_load_b64
        {
            v8f c0 = {0.f, 0.f, 0.f, 0.f, 0.f, 0.f, 0.f, 0.f};
            v8f c1 = {0.f, 0.f, 0.f, 0.f, 0.f, 0.f, 0.f, 0.f};
            const int mt  = wave & 1;
            const int ntA = (wave >> 1) * 2;
#pragma unroll
            for (int kk = 0; kk < 16; ++kk) {
                v2f a, bA, bB;
                a.x  = vp_s[(mt * 16 + lo) * PP + kk * 4 + 2 * hi + 0];
                a.y  = vp_s[(mt * 16 + lo) * PP + kk * 4 + 2 * hi + 1];
                bA.x = i_s[(ntA * 16 + lo) * IP + kk * 4 + 2 * hi + 0];
                bA.y = i_s[(ntA * 16 + lo) * IP + kk * 4 + 2 * hi + 1];
                c0 = wmma_f32(a, bA, c0);
                bB.x = i_s[((ntA + 1) * 16 + lo) * IP + kk * 4 + 2 * hi + 0];
                bB.y = i_s[((ntA + 1) * 16 + lo) * IP + kk * 4 + 2 * hi + 1];
                c1 = wmma_f32(a, bB, c1);
            }
#pragma unroll
            for (int vv = 0; vv < 8; ++vv) {
                z_s[(mt * 16 + vv + 8 * hi) * ZP + ntA * 16 + lo]       = c0[vv];
                z_s[(mt * 16 + vv + 8 * hi) * ZP + (ntA + 1) * 16 + lo] = c1[vv];
            }
        }
        __syncthreads();

        // ========== softmax over q (len 128) per s-row, then E2 = exp(9*p)
        // 8 lanes per row (row = t>>3), 16 cols each; reduce via shfl_xor 1,2,4
        {
            const int row = t >> 3;
            float* zr = &z_s[row * ZP + (t & 7) * 16];
            float m = -3.4e38f;
#pragma unroll
            for (int c = 0; c < 16; ++c) m = fmaxf(m, zr[c]);
            m = fmaxf(m, __shfl_xor(m, 1));
            m = fmaxf(m, __shfl_xor(m, 2));
            m = fmaxf(m, __shfl_xor(m, 4));
            float s = 0.0f;
#pragma unroll
            for (int c = 0; c < 16; ++c) { float e = __expf(zr[c] - m); zr[c] = e; s += e; }
            s += __shfl_xor(s, 1);
            s += __shfl_xor(s, 2);
            s += __shfl_xor(s, 4);
            const float inv = SMOOTHF / s;    // fold SMOOTH into normalization
#pragma unroll
            for (int c = 0; c < 16; ++c) zr[c] = __expf(zr[c] * inv);  // in (1, e^9]
        }
        __syncthreads();

        // ========== bucket accumulate: bucket(s0+j) = (blk&3)*32 + j
        {
            const int qb = (blk & 3) * 32;
            if ((qb >> 6) == qhalf) {
                const float* col = &z_s[l_own];
                if (blk & 1) {
#pragma unroll
                    for (int j = 0; j < 32; ++j) nb[32 + j] += col[j * ZP];
                } else {
#pragma unroll
                    for (int j = 0; j < 32; ++j) nb[j] += col[j * ZP];
                }
            }
        }
        // ========== Z2[l] += column sums of E2 block
        if (t < LQ) {
            float s = 0.0f;
#pragma unroll
            for (int j = 0; j < 32; ++j) s += z_s[j * ZP + t];
            Z2[t] += s;
        }
        __syncthreads();
    }

    // ========== w[q] = sum_l nb[l][q] / Z2[l]   (ds_add_f32 reduction over l)
    {
        const float invz = 1.0f / Z2[l_own];
#pragma unroll
        for (int k = 0; k < 64; ++k)
            atomicAdd(&w_s[qhalf * 64 + k], nb[k] * invz);
    }
    __syncthreads();

    // ========== y[bh][d] = sum_q w[q] * i[bh,q,d]
    if (t < DK) {
        float s = 0.0f;
#pragma unroll 8
        for (int q = 0; q < LQ; ++q) s += w_s[q] * i_s[q * IP + t];
        y[bh * DK + t] = s;
    }
}

// out(8,1024) = x(8,1024) @ Wo(1024,1024) + bo   (memory-bound on Wo: read once)
__global__ __launch_bounds__(256) void out_proj_kernel(
    const float* __restrict__ x,   // (8,1024) == y buffer
    const float* __restrict__ Wo,
    const float* __restrict__ bo,
    float* __restrict__ out)       // (8,1,1024)
{
    __shared__ float x_s[NBATCH * DMODEL];
    const int t = threadIdx.x;
#pragma unroll
    for (int it = 0; it < 8; ++it) {           // 2048 float4s / 256 threads
        const int idx = t + it * 256;
        async_g2l_b128(&x_s[idx * 4], &x[idx * 4]);
    }
    wait_async0();
    __syncthreads();

    const int n = blockIdx.x * 256 + t;
    float acc[NBATCH];
#pragma unroll
    for (int b = 0; b < NBATCH; ++b) acc[b] = 0.0f;
    for (int k = 0; k < DMODEL; ++k) {
        const float w = Wo[(size_t)k * DMODEL + n];
#pragma unroll
        for (int b = 0; b < NBATCH; ++b) acc[b] += x_s[b * DMODEL + k] * w;
    }
    const float bb = bo[n];
#pragma unroll
    for (int b = 0; b < NBATCH; ++b) out[(size_t)b * DMODEL + n] = acc[b] + bb;
}

extern "C" void kernel_launch(void* const* d_in, const int* in_sizes, int n_in,
                              void* d_out, int out_size, void* d_ws, size_t ws_size,
                              hipStream_t stream) {
    // setup_inputs order: q,k,v,i,Wq,bq,Wk,bk,Wv,bv,Wo,bo  (q/k/Wq/bq/Wk/bk are dead)
    const float* v  = (const float*)d_in[2];
    const float* ii = (const float*)d_in[3];
    const float* Wv = (const float*)d_in[8];
    const float* bv = (const float*)d_in[9];
    const float* Wo = (const float*)d_in[10];
    const float* bo = (const float*)d_in[11];
    float* out = (float*)d_out;

    float* y = (float*)d_ws;  // (128,64) = 32 KB scratch

    attn_fused_kernel<<<BH, 256, 0, stream>>>(v, ii, Wv, bv, y);
    out_proj_kernel<<<DMODEL / 256, 256, 0, stream>>>(y, Wo, bo, out);
}